// MolecularGCN_31361851196181
// MI455X (gfx1250) — compile-verified
//
#include <hip/hip_runtime.h>
#include <hip/hip_bf16.h>

typedef __attribute__((ext_vector_type(2))) float v2f;
typedef __attribute__((ext_vector_type(8))) float v8f;

#define F1 64
#define HID 128
#define H2 64
#define L1O 32
#define OUTD 10

// ---------------------------------------------------------------------------
// Degree / normalization
// ---------------------------------------------------------------------------
__global__ void deg_init_kernel(float* __restrict__ deg, int n) {
    int i = blockIdx.x * blockDim.x + threadIdx.x;
    if (i < n) deg[i] = 1.0f;  // self-loop contribution
}

__global__ void deg_edges_kernel(float* __restrict__ deg,
                                 const int* __restrict__ dst, int e) {
    int i = blockIdx.x * blockDim.x + threadIdx.x;
    if (i < e) atomicAdd(&deg[dst[i]], 1.0f);
}

__global__ void dinv_kernel(float* __restrict__ dinv, int n) {
    int i = blockIdx.x * blockDim.x + threadIdx.x;
    if (i < n) dinv[i] = rsqrtf(dinv[i]);  // deg >= 1 always (self loops)
}

// ---------------------------------------------------------------------------
// WMMA f32 GEMM:  hs[row][col] = dinv[row] * sum_k X[row][k] * W[k][col]
// One wave per 16x16 output tile, K-loop fully unrolled over
// V_WMMA_F32_16X16X4_F32.  W is staged *transposed* in LDS once per block so
// the B fragment is a single contiguous ds_load_b64 (DScnt) while the A
// fragment is a global_load_b64 (LOADcnt) -- independent counters overlap.
// Nrows must be a multiple of 16 (50000 = 3125*16).
// ---------------------------------------------------------------------------
template <int K, int M>
__global__ __launch_bounds__(256) void gcn_wmma_gemm_kernel(
    const float* __restrict__ X, const float* __restrict__ W,
    const float* __restrict__ dinv, float* __restrict__ HS,
    int totalTiles)
{
    __shared__ float tW[M * K];  // transposed: tW[col*K + k]

    // cooperative transposed load of W (read coalesced along m)
    for (int idx = threadIdx.x; idx < K * M; idx += 256) {
        int k = idx / M;
        int m = idx - k * M;
        tW[m * K + k] = W[idx];
    }
    __syncthreads();

    const int lane = threadIdx.x & 31;
    const int wave = blockIdx.x * 8 + (threadIdx.x >> 5);
    if (wave >= totalTiles) return;            // wave-uniform exit (EXEC all-1 for WMMA)

    constexpr int tilesN = M / 16;
    const int tileM = wave / tilesN;
    const int tileN = wave - tileM * tilesN;
    const int half  = lane >> 4;               // 0: K=k0..k0+1, 1: K=k0+2..k0+3
    const int r     = lane & 15;
    const unsigned rowA = tileM * 16 + r;
    const unsigned colB = tileN * 16 + r;

    const float* __restrict__ arow = X + rowA * (unsigned)K + 2u * half;
    const float* __restrict__ bcol = tW + colB * (unsigned)K + 2u * half;

    v8f c = {};
#pragma unroll
    for (int k0 = 0; k0 < K; k0 += 4) {
        v2f a = *(const v2f*)(arow + k0);      // global_load_b64
        v2f b = *(const v2f*)(bcol + k0);      // ds_load_b64
        c = __builtin_amdgcn_wmma_f32_16x16x4_f32(false, a, false, b,
                                                  (short)0, c, false, false);
    }
    // D layout: VGPR i -> M = i (lanes 0-15), M = i+8 (lanes 16-31); N = lane&15
#pragma unroll
    for (int i = 0; i < 8; ++i) {
        unsigned row = tileM * 16 + i + 8 * half;
        HS[row * (unsigned)M + tileN * 16 + r] = c[i] * dinv[row];
    }
}

// ---------------------------------------------------------------------------
// zero fill (float4)
// ---------------------------------------------------------------------------
__global__ void zero4_kernel(float4* __restrict__ p, unsigned n4) {
    unsigned i = blockIdx.x * blockDim.x + threadIdx.x;
    if (i < n4) p[i] = make_float4(0.f, 0.f, 0.f, 0.f);
}

// ---------------------------------------------------------------------------
// Edge scatter: acc[dst] += hs[src]   (float4 gather + 4x f32 atomics)
// thread idx -> (edge, 16B chunk); chunkShift = log2(M/4)
// ---------------------------------------------------------------------------
__global__ void gcn_scatter_kernel(const float4* __restrict__ hs,
                                   const int* __restrict__ src,
                                   const int* __restrict__ dst,
                                   float* __restrict__ acc,
                                   unsigned total, int M, int chunkShift)
{
    unsigned idx = blockIdx.x * blockDim.x + threadIdx.x;
    if (idx >= total) return;
    unsigned e  = idx >> chunkShift;
    unsigned ch = idx & ((1u << chunkShift) - 1u);
    unsigned s = (unsigned)src[e];
    unsigned d = (unsigned)dst[e];
    const float4 v = hs[s * (unsigned)(M >> 2) + ch];
    float* out = acc + d * (unsigned)M + ch * 4u;
    atomicAdd(out + 0, v.x);
    atomicAdd(out + 1, v.y);
    atomicAdd(out + 2, v.z);
    atomicAdd(out + 3, v.w);
}

// ---------------------------------------------------------------------------
// Finalize (in place on acc): acc = relu(dinv[row] * (acc + hs) + bias)
// hs term = self-loop (hs already carries one dinv factor).  float4 per thread.
// chunkShift = log2(M/4)
// ---------------------------------------------------------------------------
__global__ void gcn_finalize4_kernel(float4* __restrict__ acc,
                                     const float4* __restrict__ hs,
                                     const float* __restrict__ dinv,
                                     const float4* __restrict__ bias,
                                     unsigned total4, int chunkShift)
{
    unsigned i = blockIdx.x * blockDim.x + threadIdx.x;
    if (i >= total4) return;
    unsigned row = i >> chunkShift;
    unsigned ch  = i & ((1u << chunkShift) - 1u);
    float4 a = acc[i];
    float4 h = hs[i];
    float4 b = bias[ch];
    float dv = dinv[row];
    float4 r;
    r.x = fmaxf(dv * (a.x + h.x) + b.x, 0.0f);
    r.y = fmaxf(dv * (a.y + h.y) + b.y, 0.0f);
    r.z = fmaxf(dv * (a.z + h.z) + b.z, 0.0f);
    r.w = fmaxf(dv * (a.w + h.w) + b.w, 0.0f);
    acc[i] = r;
}

// ---------------------------------------------------------------------------
// Global mean pool accumulation: sums[batch[i]] += h[i]; cnt[batch[i]] += 1
// thread -> (node, float4 chunk of H2=64 feats -> 16 chunks)
// ---------------------------------------------------------------------------
__global__ void pool_accum_kernel(const float4* __restrict__ h,
                                  const int* __restrict__ batch,
                                  float* __restrict__ sums,
                                  float* __restrict__ cnt, unsigned n)
{
    unsigned idx = blockIdx.x * blockDim.x + threadIdx.x;
    if (idx >= n * 16u) return;
    unsigned node = idx >> 4;
    unsigned ch = idx & 15u;
    unsigned b = (unsigned)batch[node];
    const float4 v = h[node * (H2 / 4u) + ch];
    float* out = sums + b * (unsigned)H2 + ch * 4u;
    atomicAdd(out + 0, v.x);
    atomicAdd(out + 1, v.y);
    atomicAdd(out + 2, v.z);
    atomicAdd(out + 3, v.w);
    if (ch == 0) atomicAdd(&cnt[b], 1.0f);
}

// ---------------------------------------------------------------------------
// MLP head
// ---------------------------------------------------------------------------
__global__ void mlp1_kernel(const float* __restrict__ sums,
                            const float* __restrict__ cnt,
                            const float* __restrict__ Wl1,
                            const float* __restrict__ bl1,
                            float* __restrict__ g1, int B)
{
    int idx = blockIdx.x * blockDim.x + threadIdx.x;
    if (idx >= B * L1O) return;
    unsigned b = (unsigned)idx >> 5;   // L1O == 32
    unsigned j = (unsigned)idx & 31u;
    float inv = 1.0f / fmaxf(cnt[b], 1.0f);
    float s = bl1[j];
#pragma unroll
    for (int k = 0; k < H2; ++k)
        s += (sums[b * H2 + k] * inv) * Wl1[k * L1O + j];
    g1[b * L1O + j] = fmaxf(s, 0.0f);
}

__global__ void mlp2_kernel(const float* __restrict__ g1,
                            const float* __restrict__ Wl2,
                            const float* __restrict__ bl2,
                            float* __restrict__ out, int B)
{
    int idx = blockIdx.x * blockDim.x + threadIdx.x;
    if (idx >= B * OUTD) return;
    unsigned b = (unsigned)idx / OUTD;
    unsigned o = (unsigned)idx - b * OUTD;
    float s = bl2[o];
#pragma unroll
    for (int j = 0; j < L1O; ++j)
        s += g1[b * L1O + j] * Wl2[j * OUTD + o];
    out[b * OUTD + o] = s;
}

// ---------------------------------------------------------------------------
// Host driver
// ---------------------------------------------------------------------------
static inline int cdiv(long long a, int b) { return (int)((a + b - 1) / b); }

extern "C" void kernel_launch(void* const* d_in, const int* in_sizes, int n_in,
                              void* d_out, int out_size, void* d_ws, size_t ws_size,
                              hipStream_t stream) {
    const float* x     = (const float*)d_in[0];
    const int*   ei    = (const int*)d_in[1];
    const int*   batch = (const int*)d_in[2];
    const float* W1 = (const float*)d_in[3];  const float* b1 = (const float*)d_in[4];
    const float* W2 = (const float*)d_in[5];  const float* b2 = (const float*)d_in[6];
    const float* W3 = (const float*)d_in[7];  const float* b3 = (const float*)d_in[8];
    const float* Wl1 = (const float*)d_in[9]; const float* bl1 = (const float*)d_in[10];
    const float* Wl2 = (const float*)d_in[11];const float* bl2 = (const float*)d_in[12];

    const int N = in_sizes[0] / F1;      // 50000
    const int E = in_sizes[1] / 2;       // 1600000
    const int B = out_size / OUTD;       // 1000
    const int* src = ei;
    const int* dst = ei + E;

    float* ws   = (float*)d_ws;
    float* buf0 = ws;                                   // N*128
    float* buf1 = ws + (size_t)N * HID;                 // N*128
    float* buf2 = ws + 2 * (size_t)N * HID;             // N*128
    float* dinv = ws + 3 * (size_t)N * HID;             // N
    float* sums = dinv + N;                             // B*64
    float* cnt  = sums + (size_t)B * H2;                // B
    float* g1   = cnt + B;                              // B*32

    const int T = 256;

    // --- degree + dinv ---
    deg_init_kernel<<<cdiv(N, T), T, 0, stream>>>(dinv, N);
    deg_edges_kernel<<<cdiv(E, T), T, 0, stream>>>(dinv, dst, E);
    dinv_kernel<<<cdiv(N, T), T, 0, stream>>>(dinv, N);

    const int tilesM = N / 16;  // 3125

    // --- Layer 1: x(N,64) @ W1(64,128) ---
    {
        const int M = HID;
        int tiles = tilesM * (M / 16);
        gcn_wmma_gemm_kernel<F1, HID><<<cdiv(tiles, 8), 256, 0, stream>>>(
            x, W1, dinv, buf0, tiles);
        unsigned nm4 = (unsigned)N * (M / 4);
        zero4_kernel<<<cdiv(nm4, T), T, 0, stream>>>((float4*)buf1, nm4);
        unsigned tot = (unsigned)E * (M / 4);
        gcn_scatter_kernel<<<cdiv(tot, T), T, 0, stream>>>(
            (const float4*)buf0, src, dst, buf1, tot, M, 5);
        gcn_finalize4_kernel<<<cdiv(nm4, T), T, 0, stream>>>(
            (float4*)buf1, (const float4*)buf0, dinv, (const float4*)b1, nm4, 5);
    }
    // --- Layer 2: h(N,128) @ W2(128,128) ---
    {
        const int M = HID;
        int tiles = tilesM * (M / 16);
        gcn_wmma_gemm_kernel<HID, HID><<<cdiv(tiles, 8), 256, 0, stream>>>(
            buf1, W2, dinv, buf0, tiles);
        unsigned nm4 = (unsigned)N * (M / 4);
        zero4_kernel<<<cdiv(nm4, T), T, 0, stream>>>((float4*)buf2, nm4);
        unsigned tot = (unsigned)E * (M / 4);
        gcn_scatter_kernel<<<cdiv(tot, T), T, 0, stream>>>(
            (const float4*)buf0, src, dst, buf2, tot, M, 5);
        gcn_finalize4_kernel<<<cdiv(nm4, T), T, 0, stream>>>(
            (float4*)buf2, (const float4*)buf0, dinv, (const float4*)b2, nm4, 5);
    }
    // --- Layer 3: h(N,128) @ W3(128,64) ---
    {
        const int M = H2;
        int tiles = tilesM * (M / 16);
        gcn_wmma_gemm_kernel<HID, H2><<<cdiv(tiles, 8), 256, 0, stream>>>(
            buf2, W3, dinv, buf0, tiles);
        unsigned nm4 = (unsigned)N * (M / 4);
        zero4_kernel<<<cdiv(nm4, T), T, 0, stream>>>((float4*)buf1, nm4);
        unsigned tot = (unsigned)E * (M / 4);
        gcn_scatter_kernel<<<cdiv(tot, T), T, 0, stream>>>(
            (const float4*)buf0, src, dst, buf1, tot, M, 4);
        gcn_finalize4_kernel<<<cdiv(nm4, T), T, 0, stream>>>(
            (float4*)buf1, (const float4*)buf0, dinv, (const float4*)b3, nm4, 4);
    }

    // --- global mean pool + MLP head ---
    unsigned poolz4 = (unsigned)(B * H2 + B + 3) / 4;   // sums and cnt contiguous
    zero4_kernel<<<cdiv(poolz4, T), T, 0, stream>>>((float4*)sums, poolz4);
    pool_accum_kernel<<<cdiv((long long)N * 16, T), T, 0, stream>>>(
        (const float4*)buf1, batch, sums, cnt, (unsigned)N);
    mlp1_kernel<<<cdiv((long long)B * L1O, T), T, 0, stream>>>(sums, cnt, Wl1, bl1, g1, B);
    mlp2_kernel<<<cdiv((long long)B * OUTD, T), T, 0, stream>>>(g1, Wl2, bl2, (float*)d_out, B);
}